// BDHNeuroRefTorch_48601849921694
// MI455X (gfx1250) — compile-verified
//
#include <hip/hip_runtime.h>

#define NN   2048
#define DD   64
#define TT   512
#define KWTA 256

typedef float v2f __attribute__((ext_vector_type(2)));
typedef float v8f __attribute__((ext_vector_type(8)));

// ---------- bf16 helpers (manual, round-to-nearest-even) ----------
__device__ __forceinline__ float bflo(unsigned u) { return __uint_as_float(u << 16); }
__device__ __forceinline__ float bfhi(unsigned u) { return __uint_as_float(u & 0xffff0000u); }
__device__ __forceinline__ unsigned f2bf(float f) {
    unsigned u = __float_as_uint(f);
    return (u + 0x7fffu + ((u >> 16) & 1u)) >> 16;
}

// ---------- wave32 reductions ----------
__device__ __forceinline__ float wsum(float v) {
#pragma unroll
    for (int m = 16; m > 0; m >>= 1) v += __shfl_xor(v, m, 32);
    return v;
}
__device__ __forceinline__ int wsumi(int v) {
#pragma unroll
    for (int m = 16; m > 0; m >>= 1) v += __shfl_xor(v, m, 32);
    return v;
}
__device__ __forceinline__ float wmaxf(float v) {
#pragma unroll
    for (int m = 16; m > 0; m >>= 1) v = fmaxf(v, __shfl_xor(v, m, 32));
    return v;
}

__device__ __forceinline__ float softplusf(float z) {
    return (z > 20.f) ? z : log1pf(expf(z));
}

// =====================================================================
// Kernel 1: repack Dy (N,D) and E (D,N) into interleaved bf16 per column:
//   pack[n*128 + d]      = bf16(Dy[n,d])
//   pack[n*128 + 64 + d] = bf16(E[d,n])
// =====================================================================
__global__ __launch_bounds__(128) void pack_dye(const float* __restrict__ Dy,
                                                const float* __restrict__ E,
                                                unsigned short* __restrict__ pack) {
    int n = blockIdx.x;
    int i = threadIdx.x;  // 0..127
    float v = (i < 64) ? Dy[n * 64 + i] : E[(size_t)(i - 64) * NN + n];
    pack[(size_t)n * 128 + i] = (unsigned short)f2bf(v);
}

// =====================================================================
// Kernel 2: G[t,n] = relu( sum_d Emb[tokens[t],d] * Dx[n,d] )
// FP32 GEMM (512 x 2048 x 64) via v_wmma_f32_16x16x4_f32.
// One 16x16 output tile per wave; 4096 tiles total.
// =====================================================================
__global__ __launch_bounds__(128) void gemm_g(const int* __restrict__ tokens,
                                              const float* __restrict__ emb,
                                              const float* __restrict__ Dx,
                                              float* __restrict__ G) {
    int lane = threadIdx.x & 31;
    int wid  = threadIdx.x >> 5;
    int tile = blockIdx.x * 4 + wid;          // 0..4095
    int tn = (tile & 127) * 16;               // N tile base
    int tm = (tile >> 7) * 16;                // T tile base
    int half = lane >> 4;                     // 0: K even pair, 1: K odd pair (+2)
    int l    = lane & 15;

    int tokid = tokens[tm + l];
    const float* arow = emb + (size_t)tokid * DD;  // A[M=l, k]
    const float* brow = Dx + (size_t)(tn + l) * DD; // B[K][N=l] = Dx[N,K]

    v8f c = {0.f, 0.f, 0.f, 0.f, 0.f, 0.f, 0.f, 0.f};
#pragma unroll
    for (int k0 = 0; k0 < DD; k0 += 4) {
        // A 16x4 layout: lanes 0-15 hold K=k0,k0+1 ; lanes 16-31 hold K=k0+2,k0+3
        v2f a; a.x = arow[k0 + 2 * half];     a.y = arow[k0 + 2 * half + 1];
        // B 4x16 layout mirrors A: VGPR0 = K row (k0 / k0+2), VGPR1 = +1
        v2f b; b.x = brow[k0 + 2 * half];     b.y = brow[k0 + 2 * half + 1];
        c = __builtin_amdgcn_wmma_f32_16x16x4_f32(false, a, false, b,
                                                  (short)0, c, false, false);
    }
    // C layout: VGPR j holds row M=j (lanes 0-15) and M=8+j (lanes 16-31), N=l
#pragma unroll
    for (int j = 0; j < 8; ++j) {
        int row = tm + j + half * 8;
        G[(size_t)row * NN + tn + l] = fmaxf(c[j], 0.f);
    }
}

// =====================================================================
// Kernel 3: persistent sequential scan. 1 workgroup, 1024 threads.
// Thread tid owns columns n0=2*tid, n1=2*tid+1. rho is bf16 in LDS.
// =====================================================================
__global__ __launch_bounds__(1024) void bdh_seq(const int* __restrict__ tokens,
                                                const float* __restrict__ spike_u,
                                                const float* __restrict__ emb,
                                                const float* __restrict__ G,
                                                const unsigned short* __restrict__ pack,
                                                float* __restrict__ out) {
    extern __shared__ char smem[];
    unsigned* rho32 = (unsigned*)smem;                       // 64*1024 u32 = 256 KB (bf16 pairs)
    float* vprev = (float*)(smem + 64 * 1024 * 4);           // 64
    float* avec  = vprev + 64;                               // 64
    float* vvec  = avec + 64;                                // 64
    float* wpart = vvec + 64;                                // 32*64 = 2048
    float* wbuf  = wpart + 2048;                             // 32
    float* slots = wbuf + 32;                                // 12
    int*   icnt  = (int*)(slots + 12);                       // 40

    const int tid  = threadIdx.x;
    const int lane = tid & 31;
    const int wid  = tid >> 5;
    const int n0 = 2 * tid, n1 = 2 * tid + 1;

    // ---- init state ----
#pragma unroll 4
    for (int i = tid; i < 64 * 1024; i += 1024) rho32[i] = 0u;
    float x0 = 0.f, x1 = 0.f;
    __syncthreads();

    const float invlogD = 1.0f / logf((float)DD);

    for (int t = 0; t < TT; ++t) {
        // ---- zero per-step accumulators ----
        __syncthreads();
        if (tid < 64) { avec[tid] = 0.f; vvec[tid] = 0.f; }
        if (tid < 12) slots[tid] = 0.f;
        if (tid < 40) icnt[tid] = 0;
        int tok = tokens[t];
        if (tid < 64) vprev[tid] = emb[(size_t)tok * DD + tid];
        __syncthreads();

        // ---- x recurrence + L1 normalize (G already relu'd) ----
        x0 = 0.97f * x0 + G[(size_t)t * NN + n0];
        x1 = 0.97f * x1 + G[(size_t)t * NN + n1];
        {
            float p = wsum(x0 + x1);                 // x >= 0
            if (lane == 0) atomicAdd(&slots[0], p);
            __syncthreads();
            float inv = 1.0f / (slots[0] + 1e-6f);
            x0 *= inv; x1 *= inv;
            __syncthreads();
        }

        // ---- a_raw = rho @ x  (bf16 LDS, per-wave partials) ----
#pragma unroll
        for (int d = 0; d < DD; ++d) {
            unsigned rr = rho32[d * 1024 + tid];
            float p = bflo(rr) * x0 + bfhi(rr) * x1;
            p = wsum(p);
            if (lane == 0) wpart[wid * 64 + d] = p;
        }
        __syncthreads();
        if (tid < 64) {
            float a = 0.f;
            for (int w = 0; w < 32; ++w) a += wpart[w * 64 + tid];
            avec[tid] = a;
        }
        __syncthreads();

        // ---- ln_row(a): mean + unbiased std ----
        float meanA, rinvA;
        {
            float val = (tid < 64) ? avec[tid] : 0.f;
            float p = wsum(val);
            if (lane == 0) atomicAdd(&slots[1], p);
            p = wsum(val * val);
            if (lane == 0) atomicAdd(&slots[2], p);
            __syncthreads();
            meanA = slots[1] * (1.0f / 64.0f);
            float var = (slots[2] - 64.0f * meanA * meanA) * (1.0f / 63.0f);
            rinvA = 1.0f / (sqrtf(fmaxf(var, 0.f)) + 1e-6f);
            __syncthreads();
        }

        // ---- y = softplus(Dy @ a_hat) + spike ----
        float y0, y1;
        {
            const uint4* p0 = (const uint4*)(pack + (size_t)n0 * 128);
            const uint4* p1 = (const uint4*)(pack + (size_t)n1 * 128);
            float z0 = 0.f, z1 = 0.f;
#pragma unroll
            for (int i = 0; i < 8; ++i) {
                uint4 q0 = p0[i], q1 = p1[i];
                int d = i * 8;
                float ah0 = (avec[d + 0] - meanA) * rinvA;
                float ah1 = (avec[d + 1] - meanA) * rinvA;
                float ah2 = (avec[d + 2] - meanA) * rinvA;
                float ah3 = (avec[d + 3] - meanA) * rinvA;
                float ah4 = (avec[d + 4] - meanA) * rinvA;
                float ah5 = (avec[d + 5] - meanA) * rinvA;
                float ah6 = (avec[d + 6] - meanA) * rinvA;
                float ah7 = (avec[d + 7] - meanA) * rinvA;
                z0 += bflo(q0.x) * ah0 + bfhi(q0.x) * ah1 + bflo(q0.y) * ah2 + bfhi(q0.y) * ah3
                    + bflo(q0.z) * ah4 + bfhi(q0.z) * ah5 + bflo(q0.w) * ah6 + bfhi(q0.w) * ah7;
                z1 += bflo(q1.x) * ah0 + bfhi(q1.x) * ah1 + bflo(q1.y) * ah2 + bfhi(q1.y) * ah3
                    + bflo(q1.z) * ah4 + bfhi(q1.z) * ah5 + bflo(q1.w) * ah6 + bfhi(q1.w) * ah7;
            }
            y0 = softplusf(z0) + ((spike_u[(size_t)t * NN + n0] < 0.01f) ? 1.f : 0.f);
            y1 = softplusf(z1) + ((spike_u[(size_t)t * NN + n1] < 0.01f) ? 1.f : 0.f);
        }

        // ---- k-WTA (top-256) threshold via bisection ----
        float thrY;
        {
            float m = wmaxf(fmaxf(y0, y1));
            if (lane == 0) wbuf[wid] = m;
            __syncthreads();
            float ymax = wbuf[0];
#pragma unroll
            for (int w = 1; w < 32; ++w) ymax = fmaxf(ymax, wbuf[w]);
            __syncthreads();
            float lo = 0.f, hi = ymax * (1.0f + 1e-6f) + 1e-12f;
            for (int it = 0; it < 20; ++it) {
                float mid = 0.5f * (lo + hi);
                int c = (y0 > mid) + (y1 > mid);
                c = wsumi(c);
                if (lane == 0) atomicAdd(&icnt[it], c);
                __syncthreads();
                int total = icnt[it];
                if (total > KWTA) lo = mid; else hi = mid;
            }
            thrY = lo;
        }

        // ---- y_t = mask*y*x, homeostasis ----
        float yt0 = (y0 > thrY) ? y0 * x0 : 0.f;
        float yt1 = (y1 > thrY) ? y1 * x1 : 0.f;
        {
            float p = wsum(yt0 + yt1);
            if (lane == 0) atomicAdd(&slots[3], p);
            __syncthreads();
            float s = slots[3];
            float scale = (s > 1e-8f) ? fminf(1.0f, 1.0f / (s + 1e-8f)) : 1.0f;
            yt0 *= scale; yt1 *= scale;
            __syncthreads();
        }

        // ---- v_raw = E @ y_t (bf16 packed, per-wave partials) ----
        {
            const unsigned* pe0 = (const unsigned*)(pack + (size_t)n0 * 128 + 64);
            const unsigned* pe1 = (const unsigned*)(pack + (size_t)n1 * 128 + 64);
#pragma unroll
            for (int h = 0; h < 32; ++h) {
                unsigned w0 = pe0[h], w1 = pe1[h];
                float p = bflo(w0) * yt0 + bflo(w1) * yt1;
                p = wsum(p);
                if (lane == 0) wpart[wid * 64 + 2 * h] = p;
                p = bfhi(w0) * yt0 + bfhi(w1) * yt1;
                p = wsum(p);
                if (lane == 0) wpart[wid * 64 + 2 * h + 1] = p;
            }
            __syncthreads();
            if (tid < 64) {
                float a = 0.f;
                for (int w = 0; w < 32; ++w) a += wpart[w * 64 + tid];
                vvec[tid] = a;
            }
            __syncthreads();
        }

        // ---- ln_row(v) -> v_star, write output ----
        float meanV, rinvV;
        {
            float val = (tid < 64) ? vvec[tid] : 0.f;
            float p = wsum(val);
            if (lane == 0) atomicAdd(&slots[4], p);
            p = wsum(val * val);
            if (lane == 0) atomicAdd(&slots[5], p);
            __syncthreads();
            meanV = slots[4] * (1.0f / 64.0f);
            float var = (slots[5] - 64.0f * meanV * meanV) * (1.0f / 63.0f);
            rinvV = 1.0f / (sqrtf(fmaxf(var, 0.f)) + 1e-6f);
            if (tid < 64) out[(size_t)t * DD + tid] = (vvec[tid] - meanV) * rinvV;
            __syncthreads();
        }

        // ---- entropy of softmax(v_star) -> gain ----
        float gain;
        {
            float vs = (tid < 64) ? (vvec[tid] - meanV) * rinvV : -1e30f;
            float m = wmaxf(vs);
            if (lane == 0) wbuf[wid] = m;
            __syncthreads();
            float vmax = wbuf[0];
#pragma unroll
            for (int w = 1; w < 32; ++w) vmax = fmaxf(vmax, wbuf[w]);
            __syncthreads();
            float e = (tid < 64) ? expf(vs - vmax) : 0.f;
            float p = wsum(e);
            if (lane == 0) atomicAdd(&slots[6], p);
            p = wsum((tid < 64) ? e * vs : 0.f);
            if (lane == 0) atomicAdd(&slots[7], p);
            __syncthreads();
            float sumexp = slots[6];
            float H = (vmax + logf(sumexp)) - slots[7] / sumexp;
            gain = fminf(1.0f, H * invlogD);
            __syncthreads();
        }
        int k_dyn = (int)((float)NN * (0.05f + 0.25f * gain));
        k_dyn = max(1, min(NN, k_dyn));

        // ---- dynamic top-k mask on x via bisection ----
        {
            float m = wmaxf(fmaxf(x0, x1));
            if (lane == 0) wbuf[wid] = m;
            __syncthreads();
            float xmax = wbuf[0];
#pragma unroll
            for (int w = 1; w < 32; ++w) xmax = fmaxf(xmax, wbuf[w]);
            __syncthreads();
            float lo = 0.f, hi = xmax * (1.0f + 1e-6f) + 1e-12f;
            for (int it = 0; it < 16; ++it) {
                float mid = 0.5f * (lo + hi);
                int c = (x0 > mid) + (x1 > mid);
                c = wsumi(c);
                if (lane == 0) atomicAdd(&icnt[20 + it], c);
                __syncthreads();
                int total = icnt[20 + it];
                if (total > k_dyn) lo = mid; else hi = mid;
            }
            float thrX = lo;
            x0 = (x0 > thrX) ? x0 : 0.f;
            x1 = (x1 > thrX) ? x1 : 0.f;
        }

        // ---- rho: forget inactive cols, Hebbian outer-product, decay ----
        float fac0 = (x0 <= 0.f) ? 0.99f : 1.0f;
        float fac1 = (x1 <= 0.f) ? 0.99f : 1.0f;
#pragma unroll
        for (int d = 0; d < DD; ++d) {
            unsigned rr = rho32[d * 1024 + tid];
            float g = gain * vprev[d];
            float r0 = 0.97f * (bflo(rr) * fac0 + g * x0);
            float r1 = 0.97f * (bfhi(rr) * fac1 + g * x1);
            rho32[d * 1024 + tid] = f2bf(r0) | (f2bf(r1) << 16);
        }
        // step-start __syncthreads() orders rho writes vs next matvec reads
    }
}

// =====================================================================
extern "C" void kernel_launch(void* const* d_in, const int* in_sizes, int n_in,
                              void* d_out, int out_size, void* d_ws, size_t ws_size,
                              hipStream_t stream) {
    (void)in_sizes; (void)n_in; (void)out_size; (void)ws_size;
    const int*   tokens  = (const int*)d_in[0];
    const float* spike_u = (const float*)d_in[1];
    const float* E       = (const float*)d_in[2];
    const float* Dx      = (const float*)d_in[3];
    const float* Dy      = (const float*)d_in[4];
    const float* emb     = (const float*)d_in[5];
    float* out = (float*)d_out;

    unsigned short* pack = (unsigned short*)d_ws;                 // 2048*128*2 = 512 KB
    float* G = (float*)((char*)d_ws + (size_t)NN * 128 * 2);      // 512*2048*4 = 4 MB

    pack_dye<<<NN, 128, 0, stream>>>(Dy, E, pack);
    gemm_g<<<1024, 128, 0, stream>>>(tokens, emb, Dx, G);

    size_t smem = (size_t)64 * 1024 * 4                            // rho bf16 pairs
                + (64 + 64 + 64 + 2048 + 32 + 12) * sizeof(float)  // vectors/partials/slots
                + 40 * sizeof(int);                                // bisection counters
    bdh_seq<<<1, 1024, smem, stream>>>(tokens, spike_u, emb, G, pack, out);
}